// FullAtomReconstruction_42949672960616
// MI455X (gfx1250) — compile-verified
//
#include <hip/hip_runtime.h>

// ---------------------------------------------------------------------------
// Full-atom reconstruction (AlphaFold torsion -> atom14 coordinates)
//   N=128, L=2048, 21 restypes, 8 rigid groups, 14 atoms.
// Memory-bound (~68 MB traffic, ~0.4 GFLOP). Strategy:
//   * TDM DMA the 12.8 KB restype tables into LDS (TENSORcnt-tracked)
//   * one thread per residue, frames in registers, cndmask frame select
// ---------------------------------------------------------------------------

#define NRES 21
#define NGRP 8
#define NATM 14

#define ROT_DW   (NRES * NGRP * 9)   // 1512 dwords
#define TRN_DW   (NRES * NGRP * 3)   //  504 dwords
#define GRP_DW   (NRES * NATM)       //  294 dwords
#define POS_DW   (NRES * NATM * 3)   //  882 dwords

typedef unsigned int v4u __attribute__((ext_vector_type(4)));
typedef int          v4i __attribute__((ext_vector_type(4)));
typedef int          v8i __attribute__((ext_vector_type(8)));

// Issue one TDM load of `nd` dwords from global `gptr` into LDS offset
// `lds_off`. 1-D tensor: tile_dim0 = tensor_dim0 = nd, dim1 = 1.
__device__ __forceinline__ void tdm_load_dwords(unsigned lds_off,
                                                const void* gptr,
                                                unsigned nd) {
  unsigned long long ga = (unsigned long long)gptr;
  // D# group 0: count=1 | lds_addr | global_addr[56:0] | type=2
  v4u g0 = { 1u,
             lds_off,
             (unsigned)(ga & 0xFFFFFFFFu),
             (unsigned)((ga >> 32) & 0x01FFFFFFu) | (2u << 30) };
  // D# group 1 (bit-packed little-endian dwords):
  //  w0: workgroup_mask=0, data_size=2 (4B)
  //  w1[31:16] = tensor_dim0[15:0]
  //  w2[15:0]  = tensor_dim0[31:16], w2[31:16] = tensor_dim1[15:0] = 1
  //  w3[31:16] = tile_dim0
  //  w4[15:0]  = tile_dim1 = 1, tile_dim2 = 0
  //  w5        = tensor_dim0_stride[31:0]
  v8i g1 = { (int)(2u << 16),
             (int)((nd & 0xFFFFu) << 16),
             (int)(((nd >> 16) & 0xFFFFu) | (1u << 16)),
             (int)((nd & 0xFFFFu) << 16),
             1,
             (int)nd,
             0, 0 };
  v4i g2 = {0, 0, 0, 0};   // tensor is <= 2-D: groups 2/3 unused
  v4i g3 = {0, 0, 0, 0};
#if defined(__clang_major__) && (__clang_major__ >= 23)
  v8i g4 = {0, 0, 0, 0, 0, 0, 0, 0};
  __builtin_amdgcn_tensor_load_to_lds(g0, g1, g2, g3, g4, 0);
#else
  __builtin_amdgcn_tensor_load_to_lds(g0, g1, g2, g3, 0);
#endif
}

__device__ __forceinline__ void mat_mul3(const float* A, const float* B,
                                         float* C) {
#pragma unroll
  for (int i = 0; i < 3; ++i)
#pragma unroll
    for (int j = 0; j < 3; ++j)
      C[i * 3 + j] = A[i * 3 + 0] * B[0 * 3 + j] +
                     A[i * 3 + 1] * B[1 * 3 + j] +
                     A[i * 3 + 2] * B[2 * 3 + j];
}

// Right-multiply M (row-major 3x3) by x-axis rotation [[1,0,0],[0,c,-s],[0,s,c]]
__device__ __forceinline__ void apply_rx(float* M, float s, float c) {
#pragma unroll
  for (int i = 0; i < 3; ++i) {
    float m1 = M[i * 3 + 1], m2 = M[i * 3 + 2];
    M[i * 3 + 1] = c * m1 + s * m2;
    M[i * 3 + 2] = -s * m1 + c * m2;
  }
}

__global__ __launch_bounds__(256) void far_kernel(
    const float* __restrict__ Rbb,     // (NL,3,3)
    const float* __restrict__ tbb,     // (NL,3)
    const float* __restrict__ alpha,   // (NL,5,2)
    const int*   __restrict__ aa,      // (NL)
    const float* __restrict__ rigid_rot,    // (21,8,3,3)
    const float* __restrict__ rigid_trans,  // (21,8,3)
    const int*   __restrict__ atom14_group, // (21,14)
    const float* __restrict__ atom14_pos,   // (21,14,3)
    float* __restrict__ out,           // (NL,14,3)
    int NL) {
  __shared__ float lds_rot[ROT_DW];
  __shared__ float lds_trn[TRN_DW];
  __shared__ int   lds_grp[GRP_DW];
  __shared__ float lds_pos[POS_DW];

  // ---- stage restype tables into LDS via the Tensor Data Mover -----------
  if (threadIdx.x < 32) {  // wave 0 only issues the DMAs
    tdm_load_dwords((unsigned)(unsigned long long)(const void*)lds_rot,
                    rigid_rot, ROT_DW);
    tdm_load_dwords((unsigned)(unsigned long long)(const void*)lds_trn,
                    rigid_trans, TRN_DW);
    tdm_load_dwords((unsigned)(unsigned long long)(const void*)lds_grp,
                    atom14_group, GRP_DW);
    tdm_load_dwords((unsigned)(unsigned long long)(const void*)lds_pos,
                    atom14_pos, POS_DW);
  }
  __builtin_amdgcn_s_wait_tensorcnt(0);
  __syncthreads();

  const int idx = blockIdx.x * blockDim.x + threadIdx.x;
  if (idx >= NL) return;

  // ---- per-residue inputs -------------------------------------------------
  float R[9], t[3], al[10];
#pragma unroll
  for (int i = 0; i < 9; ++i) R[i] = Rbb[idx * 9 + i];
#pragma unroll
  for (int i = 0; i < 3; ++i) t[i] = tbb[idx * 3 + i];
#pragma unroll
  for (int i = 0; i < 10; ++i) al[i] = alpha[idx * 10 + i];
  const int a = aa[idx];

  // normalized (sin, cos) for psi, chi1..chi4
  float sn[5], cs[5];
#pragma unroll
  for (int k = 0; k < 5; ++k) {
    float x = al[2 * k], y = al[2 * k + 1];
    float inv = rsqrtf(x * x + y * y + 1e-8f);
    sn[k] = x * inv;
    cs[k] = y * inv;
  }

  const float* rot = &lds_rot[a * NGRP * 9];
  const float* trn = &lds_trn[a * NGRP * 3];
  const int*   grp = &lds_grp[a * NATM];
  const float* aps = &lds_pos[a * NATM * 3];

  // 6 distinct frames: [0]=bb (groups 0..2), [1]=psi, [2..5]=chi1..chi4
  float FR[6][9], FT[6][3];
#pragma unroll
  for (int i = 0; i < 9; ++i) FR[0][i] = R[i];
#pragma unroll
  for (int i = 0; i < 3; ++i) FT[0][i] = t[i];

  // psi frame: (bb o rigid[3]) o Rx(psi)  (translation unchanged by Rx)
  {
    float Rp[9], tp[3];
    mat_mul3(R, rot + 3 * 9, Rp);
#pragma unroll
    for (int i = 0; i < 3; ++i)
      tp[i] = R[i * 3 + 0] * trn[3 * 3 + 0] + R[i * 3 + 1] * trn[3 * 3 + 1] +
              R[i * 3 + 2] * trn[3 * 3 + 2] + t[i];
    apply_rx(Rp, sn[0], cs[0]);
#pragma unroll
    for (int i = 0; i < 9; ++i) FR[1][i] = Rp[i];
#pragma unroll
    for (int i = 0; i < 3; ++i) FT[1][i] = tp[i];
  }

  // chi chain: chi_k = chi_{k-1} o rigid[4+k] o Rx(chi_k)
  {
    float Rc[9], tc[3];
#pragma unroll
    for (int i = 0; i < 9; ++i) Rc[i] = R[i];
#pragma unroll
    for (int i = 0; i < 3; ++i) tc[i] = t[i];
#pragma unroll
    for (int k = 0; k < 4; ++k) {
      const float* rk = rot + (4 + k) * 9;
      const float* tk = trn + (4 + k) * 3;
      float Rn[9], tn[3];
      mat_mul3(Rc, rk, Rn);
#pragma unroll
      for (int i = 0; i < 3; ++i)
        tn[i] = Rc[i * 3 + 0] * tk[0] + Rc[i * 3 + 1] * tk[1] +
                Rc[i * 3 + 2] * tk[2] + tc[i];
      apply_rx(Rn, sn[1 + k], cs[1 + k]);
#pragma unroll
      for (int i = 0; i < 9; ++i) { Rc[i] = Rn[i]; FR[2 + k][i] = Rn[i]; }
#pragma unroll
      for (int i = 0; i < 3; ++i) { tc[i] = tn[i]; FT[2 + k][i] = tn[i]; }
    }
  }

  // ---- transform the 14 atoms --------------------------------------------
  float* o = out + (long long)idx * (NATM * 3);
#pragma unroll
  for (int at = 0; at < NATM; ++at) {
    int g = grp[at];
    int f = (g < 3) ? 0 : ((g == 3) ? 1 : (g - 2));  // map group -> frame id

    float Rf[9], tf[3];
#pragma unroll
    for (int j = 0; j < 9; ++j) {
      float v = FR[0][j];
#pragma unroll
      for (int ff = 1; ff < 6; ++ff) v = (f == ff) ? FR[ff][j] : v;
      Rf[j] = v;
    }
#pragma unroll
    for (int j = 0; j < 3; ++j) {
      float v = FT[0][j];
#pragma unroll
      for (int ff = 1; ff < 6; ++ff) v = (f == ff) ? FT[ff][j] : v;
      tf[j] = v;
    }

    float px = aps[at * 3 + 0], py = aps[at * 3 + 1], pz = aps[at * 3 + 2];
#pragma unroll
    for (int i = 0; i < 3; ++i)
      o[at * 3 + i] =
          Rf[i * 3 + 0] * px + Rf[i * 3 + 1] * py + Rf[i * 3 + 2] * pz + tf[i];
  }
}

extern "C" void kernel_launch(void* const* d_in, const int* in_sizes, int n_in,
                              void* d_out, int out_size, void* d_ws,
                              size_t ws_size, hipStream_t stream) {
  const float* Rbb          = (const float*)d_in[0];
  const float* tbb          = (const float*)d_in[1];
  const float* alpha        = (const float*)d_in[2];
  const int*   aa           = (const int*)d_in[3];
  const float* rigid_rot    = (const float*)d_in[4];
  const float* rigid_trans  = (const float*)d_in[5];
  const int*   atom14_group = (const int*)d_in[6];
  const float* atom14_pos   = (const float*)d_in[7];
  float*       out          = (float*)d_out;

  const int NL = in_sizes[3];  // N*L residues
  const int block = 256;
  const int grid = (NL + block - 1) / block;
  far_kernel<<<grid, block, 0, stream>>>(Rbb, tbb, alpha, aa, rigid_rot,
                                         rigid_trans, atom14_group, atom14_pos,
                                         out, NL);
}